// EulerBias_28475633173097
// MI455X (gfx1250) — compile-verified
//
#include <hip/hip_runtime.h>
#include <hip/hip_bf16.h>

// Problem constants from the reference.
#define GAMMA_F 1.4f
#define EPS_F   1e-6f
#define N_NEWTON 20
#define B_DIM  16
#define K_DIM  64
#define NT_DIM 128
#define NX_DIM 256
#define NIF    (K_DIM - 1)           // 63 interfaces per batch
#define NPTS   (NT_DIM * NX_DIM)     // 32768 points per batch

typedef __attribute__((ext_vector_type(4))) float        v4f;
typedef __attribute__((ext_vector_type(4))) unsigned int u32x4;
typedef __attribute__((ext_vector_type(8))) unsigned int u32x8;

// ---------------------------------------------------------------------------
// Phase 1: exact Riemann solver (Newton), 1008 independent lanes. Tiny.
// ---------------------------------------------------------------------------
__device__ __forceinline__ void wave_f_df(float p, float pK, float AK, float BK,
                                          float cK, float& f, float& df) {
    const float gm1 = GAMMA_F - 1.0f;                            // 0.4
    const float exp_rare = gm1 / (2.0f * GAMMA_F);               // 1/7
    const float neg_exp  = -(GAMMA_F + 1.0f) / (2.0f * GAMMA_F); // -6/7

    float denom    = fmaxf(p + BK, EPS_F);
    float sqrt_AoD = sqrtf(fmaxf(AK / denom, EPS_F));
    float f_shock  = (p - pK) * sqrt_AoD;
    float df_shock = sqrt_AoD * (1.0f - (p - pK) / (2.0f * denom));

    float pKc     = fmaxf(pK, EPS_F);
    float p_ratio = fmaxf(p / pKc, EPS_F);
    float f_rare  = (2.0f * cK / gm1) * (powf(p_ratio, exp_rare) - 1.0f);
    float df_rare = cK / (GAMMA_F * pKc) * powf(p_ratio, neg_exp);

    bool is_shock = p > pK;
    f  = is_shock ? f_shock  : f_rare;
    df = is_shock ? df_shock : df_rare;
}

__global__ void riemann_solve_kernel(const float* __restrict__ ks,
                                     const float* __restrict__ ks_v,
                                     const float* __restrict__ ks_p,
                                     float* __restrict__ sl_out,   // [B][NIF]
                                     float* __restrict__ sr_out) { // [B][NIF]
    int idx = blockIdx.x * blockDim.x + threadIdx.x;
    if (idx >= B_DIM * NIF) return;
    int b = idx / NIF;
    int i = idx - b * NIF;

    const float gm1 = GAMMA_F - 1.0f;
    const float gp1 = GAMMA_F + 1.0f;
    const float exp_rare = gm1 / (2.0f * GAMMA_F);   // 1/7
    const float gp1_o_2g = gp1 / (2.0f * GAMMA_F);   // 6/7

    const float* kb  = ks   + b * K_DIM;
    const float* kvb = ks_v + b * K_DIM;
    const float* kpb = ks_p + b * K_DIM;

    float rho_L = kb[i],  rho_R = kb[i + 1];
    float u_L   = kvb[i], u_R   = kvb[i + 1];
    float p_L   = kpb[i], p_R   = kpb[i + 1];

    float c_L = sqrtf(fmaxf(GAMMA_F * p_L / fmaxf(rho_L, EPS_F), EPS_F));
    float c_R = sqrtf(fmaxf(GAMMA_F * p_R / fmaxf(rho_R, EPS_F), EPS_F));
    float A_L = 2.0f / (gp1 * fmaxf(rho_L, EPS_F));
    float A_R = 2.0f / (gp1 * fmaxf(rho_R, EPS_F));
    float B_L = gm1 / gp1 * p_L;
    float B_R = gm1 / gp1 * p_R;

    // Two-rarefaction initial guess.
    float num  = c_L + c_R - 0.5f * gm1 * (u_R - u_L);
    float den  = c_L / powf(fmaxf(p_L, EPS_F), exp_rare)
               + c_R / powf(fmaxf(p_R, EPS_F), exp_rare);
    float p_star = fmaxf(powf(num / den, 1.0f / exp_rare), EPS_F);

    float du = u_R - u_L;
    #pragma unroll 1
    for (int it = 0; it < N_NEWTON; ++it) {
        float fL, dfL, fR, dfR;
        wave_f_df(p_star, p_L, A_L, B_L, c_L, fL, dfL);
        wave_f_df(p_star, p_R, A_R, B_R, c_R, fR, dfR);
        float residual = fL + fR + du;
        float jacobian = fmaxf(dfL + dfR, EPS_F);
        p_star = fmaxf(p_star - residual / jacobian, EPS_F);
    }

    float sig1 = u_L - c_L * sqrtf(fmaxf(1.0f + gp1_o_2g * (p_star / fmaxf(p_L, EPS_F) - 1.0f), EPS_F));
    float sig3 = u_R + c_R * sqrtf(fmaxf(1.0f + gp1_o_2g * (p_star / fmaxf(p_R, EPS_F) - 1.0f), EPS_F));
    float speed_left  = (p_star > p_L) ? sig1 : (u_L - c_L);
    float speed_right = (p_star > p_R) ? sig3 : (u_R + c_R);

    sl_out[idx] = speed_left;
    sr_out[idx] = speed_right;
}

// ---------------------------------------------------------------------------
// Tensor Data Mover: 1D row (nelem f32) Global -> LDS via one D# descriptor.
// D# group0 (4 SGPRs): {count=1, lds_addr, global_addr[56:0], type=2}
// D# group1 (8 SGPRs): {wg_mask=0, data_size=4B, tensor_dim0=nelem,
//                       tensor_dim1=1, tile_dim0=nelem, tile_dim1/2=0,
//                       tensor_dim0_stride=nelem}
// 2-group form (VADDR2/VADDR3 = NULL) is valid for tensors up to 2D.
// ---------------------------------------------------------------------------
__device__ __forceinline__ void tdm_load_row(const float* gsrc,
                                             const float* lds_dst,
                                             unsigned nelem) {
    unsigned long long ga = (unsigned long long)(uintptr_t)gsrc;
    // Low 32 bits of the generic (flat) LDS address are the LDS byte offset.
    unsigned lds_off = (unsigned)(uintptr_t)lds_dst;

    u32x4 g0;
    g0[0] = 1u;                                       // count=1, user descriptor
    g0[1] = lds_off;                                  // lds_addr
    g0[2] = (unsigned)(ga & 0xFFFFFFFFull);           // global_addr[31:0]
    g0[3] = ((unsigned)(ga >> 32) & 0x01FFFFFFu)      // global_addr[56:32]
          | (2u << 30);                               // type = 2 ("image")

    u32x8 g1;
    g1[0] = (2u << 16);                               // wg_mask=0, data_size=4B
    g1[1] = (nelem & 0xFFFFu) << 16;                  // tensor_dim0[15:0]
    g1[2] = ((nelem >> 16) & 0xFFFFu) | (1u << 16);   // tensor_dim0[31:16] | tensor_dim1=1
    g1[3] = (nelem & 0xFFFFu) << 16;                  // tile_dim0 = nelem
    g1[4] = 0u;                                       // tile_dim1=0, tile_dim2=0 (unused)
    g1[5] = nelem;                                    // tensor_dim0_stride[31:0]
    g1[6] = 0u;                                       // stride hi / dim1_stride lo
    g1[7] = 0u;                                       // dim1_stride hi

    asm volatile("tensor_load_to_lds %0, %1" :: "s"(g0), "s"(g1) : "memory");
}

// ---------------------------------------------------------------------------
// Phase 2: bias field fill. Store-bandwidth bound (134 MB out @ 23.3 TB/s).
// 16 threads per (b,t,x) point; each owns one float4 of the contiguous K axis
// -> coalesced non-temporal global_store_b128 (write-once stream).
// Per-batch LUTs staged into LDS by the TDM (wave 0 issues; TDM ops are
// per-wave and EXEC-independent), then broadcast-read via ds_load.
// ---------------------------------------------------------------------------
__global__ void bias_fill_kernel(const float* __restrict__ xs,         // [B][K+1]
                                 const float* __restrict__ pieces_mask,// [B][K]
                                 const float* __restrict__ t_coords,   // [B][NPTS]
                                 const float* __restrict__ x_coords,   // [B][NPTS]
                                 const float* __restrict__ sl,         // [B][NIF]
                                 const float* __restrict__ sr,         // [B][NIF]
                                 float* __restrict__ out) {            // [B][NPTS][K]
    __shared__ float s_xd[K_DIM];
    __shared__ float s_sl[K_DIM];
    __shared__ float s_sr[K_DIM];
    __shared__ float s_mask[K_DIM];

    const int tid = threadIdx.x;
    const int b   = blockIdx.y;

    // Wave 0 (tid<32 is wave-uniform under wave32) issues 4 TDM descriptors,
    // waits TENSORcnt==0, then the workgroup barrier releases waves 1..7.
    if (tid < 32) {
        tdm_load_row(xs + (size_t)b * (K_DIM + 1) + 1, s_xd, NIF);
        tdm_load_row(sl + (size_t)b * NIF,             s_sl, NIF);
        tdm_load_row(sr + (size_t)b * NIF,             s_sr, NIF);
        tdm_load_row(pieces_mask + (size_t)b * K_DIM,  s_mask, K_DIM);
        __builtin_amdgcn_s_wait_tensorcnt(0);
    }
    __syncthreads();

    const int gt    = blockIdx.x * blockDim.x + tid;  // [0, NPTS*16)
    const int p     = gt >> 4;                        // point within batch
    const int chunk = gt & 15;                        // which float4 of k-axis

    const float t     = t_coords[b * NPTS + p];
    const float x     = x_coords[b * NPTS + p];
    const float inv_t = 1.0f / (t + EPS_F);

    float res[4];
    const int k0 = chunk << 2;
    #pragma unroll
    for (int j = 0; j < 4; ++j) {
        const int k = k0 + j;
        float pl = 0.0f, pr = 0.0f;
        if (k < NIF)  // penalty_left_seg padded at the end
            pl = fmaxf((x - s_xd[k]) * inv_t - s_sr[k], 0.0f);
        if (k >= 1)   // penalty_right_seg padded at the front
            pr = fmaxf(s_sl[k - 1] - (x - s_xd[k - 1]) * inv_t, 0.0f);
        float bias = -(pl + pr);
        float m = s_mask[k];
        res[j] = bias * m + ((m == 0.0f) ? -1.0e9f : 0.0f);
    }

    v4f v = {res[0], res[1], res[2], res[3]};
    size_t off = ((size_t)b * NPTS + (size_t)p) * K_DIM + (size_t)k0;
    __builtin_nontemporal_store(v, (v4f*)(out + off));   // global_store_b128 th:NT
}

// ---------------------------------------------------------------------------
// Launch
// ---------------------------------------------------------------------------
extern "C" void kernel_launch(void* const* d_in, const int* in_sizes, int n_in,
                              void* d_out, int out_size, void* d_ws, size_t ws_size,
                              hipStream_t stream) {
    const float* xs          = (const float*)d_in[0];
    const float* ks          = (const float*)d_in[1];
    const float* ks_v        = (const float*)d_in[2];
    const float* ks_p        = (const float*)d_in[3];
    const float* pieces_mask = (const float*)d_in[4];
    const float* t_coords    = (const float*)d_in[5];
    const float* x_coords    = (const float*)d_in[6];
    float* out = (float*)d_out;

    // Workspace: speed_left then speed_right, each B*NIF floats (8 KB total).
    float* ws_sl = (float*)d_ws;
    float* ws_sr = ws_sl + B_DIM * NIF;

    // Phase 1: 1008 interfaces.
    riemann_solve_kernel<<<dim3(4), dim3(256), 0, stream>>>(ks, ks_v, ks_p, ws_sl, ws_sr);

    // Phase 2: B*NPTS*16 threads, 256/block -> grid (NPTS*16/256, B).
    dim3 grid((NPTS * 16) / 256, B_DIM);
    bias_fill_kernel<<<grid, dim3(256), 0, stream>>>(xs, pieces_mask, t_coords,
                                                     x_coords, ws_sl, ws_sr, out);
}